// RayTracing_39127152066675
// MI455X (gfx1250) — compile-verified
//
#include <hip/hip_runtime.h>
#include <math.h>

// ---- reference constants ----
#define OBJ_R     1.0f
#define SDF_THR   5e-5f
#define ST_ITERS  10
#define N_STEPS   100
#define N_SECANT  8
#define INV99     (1.0f / 99.0f)

struct Ray { float cx, cy, cz, dx, dy, dz, r; };

// sdf(cam + t*dir) = ||cam + t*dir|| - r   (matches reference _sdf on the ray)
__device__ __forceinline__ float sdf_t(const Ray& R, float t) {
    float px = fmaf(t, R.dx, R.cx);
    float py = fmaf(t, R.dy, R.cy);
    float pz = fmaf(t, R.dz, R.cz);
    return sqrtf(px * px + py * py + pz * pz) - R.r;
}

// z = -sl*(zh-zl)/den + zl  with den = (sh-sl)==0 ? 1 : (sh-sl)
__device__ __forceinline__ float zpred(float sl, float sh, float zl, float zh) {
    float den = sh - sl;
    den = (den == 0.0f) ? 1.0f : den;
    return (-sl * (zh - zl)) / den + zl;
}

__global__ void __launch_bounds__(256)
raytrace_kernel(const float* __restrict__ cam,
                const float* __restrict__ dirs,
                const float* __restrict__ radius_p,
                float* __restrict__ out, int M)
{
    int i = blockIdx.x * blockDim.x + threadIdx.x;
    if (i >= M) return;

    // uniform scalars (compiler lifts to s_load)
    const float cx = cam[0], cy = cam[1], cz = cam[2];
    const float r  = radius_p[0];

    // streaming per-ray direction (12B, AoS) + prefetch ahead (global_prefetch_b8)
    const float* dp = dirs + 3ull * (unsigned)i;
    int pf = i + 2048; pf = (pf < M) ? pf : (M - 1);
    __builtin_prefetch(dirs + 3ull * (unsigned)pf, 0, 0);
    const float dx = dp[0], dy = dp[1], dz = dp[2];

    const Ray R{cx, cy, cz, dx, dy, dz, r};

    // ---- sphere intersection with unit bounding sphere ----
    const float rcd = dx * cx + dy * cy + dz * cz;       // dot(dir, cam)
    const float c2  = cx * cx + cy * cy + cz * cz;       // |cam|^2
    const float under = rcd * rcd - (c2 - OBJ_R * OBJ_R);
    const bool  maskb = under > 0.0f;
    const float sq = sqrtf(maskb ? under : 1.0f);

    float acc_s = 0.0f, acc_e = 0.0f;
    if (maskb) {
        acc_s = fmaxf(-sq - rcd, 0.0f);
        acc_e = fmaxf( sq - rcd, 0.0f);
    }

    // ---- sphere tracing (fixed 10 masked iterations, exactly as reference) ----
    bool unf_s = maskb, unf_e = maskb;
    float ns = unf_s ? sdf_t(R, acc_s) : 0.0f;
    float ne = unf_e ? sdf_t(R, acc_e) : 0.0f;

    // top()
    float cs = unf_s ? ns : 0.0f; cs = (cs <= SDF_THR) ? 0.0f : cs;
    float ce = unf_e ? ne : 0.0f; ce = (ce <= SDF_THR) ? 0.0f : ce;
    unf_s = unf_s && (cs > SDF_THR);
    unf_e = unf_e && (ce > SDF_THR);

    for (int it = 0; it < ST_ITERS; ++it) {
        acc_s += cs;
        acc_e -= ce;
        ns = unf_s ? sdf_t(R, acc_s) : 0.0f;
        ne = unf_e ? sdf_t(R, acc_e) : 0.0f;
        // line-search back-off (LINE_ITERS==1, step = 0.5)
        if (ns < 0.0f) { acc_s -= 0.5f * cs; ns = sdf_t(R, acc_s); }
        if (ne < 0.0f) { acc_e += 0.5f * ce; ne = sdf_t(R, acc_e); }
        const bool ok = acc_s < acc_e;
        unf_s = unf_s && ok;
        unf_e = unf_e && ok;
        // top()
        cs = unf_s ? ns : 0.0f; cs = (cs <= SDF_THR) ? 0.0f : cs;
        ce = unf_e ? ne : 0.0f; ce = (ce <= SDF_THR) ? 0.0f : ce;
        unf_s = unf_s && (cs > SDF_THR);
        unf_e = unf_e && (ce > SDF_THR);
    }

    // defaults from sphere tracing: pts_s = cam + acc_s*dir (holds for all mask states)
    float px = fmaf(acc_s, dx, cx);
    float py = fmaf(acc_s, dy, cy);
    float pz = fmaf(acc_s, dz, cz);
    float acc_out = acc_s;
    float net_out = (acc_s < acc_e) ? 1.0f : 0.0f;

    // ---- ray sampler (only for unresolved rays; whole waves skip via execz) ----
    const bool sampler_mask = unf_s;
    if (sampler_mask) {
        const float t_min = acc_s, t_max = acc_e;
        const float dt = t_max - t_min;

        // argmin over sign(sdf) * [100..1]  -> first sign flip (first-min kept by strict <)
        float best = 3.4e38f;
        int ind = 0;
        #pragma unroll 4
        for (int s = 0; s < N_STEPS; ++s) {
            const float t  = (float)s * INV99;
            const float dd = fmaf(t, dt, t_min);          // t_min + t*(t_max - t_min)
            const float sv = sdf_t(R, dd);
            const float sg = (sv > 0.0f) ? 1.0f : ((sv < 0.0f) ? -1.0f : 0.0f);
            const float tmp = sg * (float)(N_STEPS - s);
            if (tmp < best) { best = tmp; ind = s; }
        }

        float dists  = fmaf((float)ind * INV99, dt, t_min);
        const float sdf_at = sdf_t(R, dists);
        const bool  net_surface = sdf_at < 0.0f;

        const int   ilo   = (ind == 0) ? (N_STEPS - 1) : (ind - 1);  // torch wrap
        float z_low    = fmaf((float)ilo * INV99, dt, t_min);
        float sdf_low  = sdf_t(R, z_low);
        float z_high   = dists;
        float sdf_high = sdf_at;

        float z = zpred(sdf_low, sdf_high, z_low, z_high);
        #pragma unroll
        for (int k = 0; k < N_SECANT; ++k) {
            const float smid = sdf_t(R, z);
            if (smid > 0.0f) { z_low  = z; sdf_low  = smid; }
            if (smid < 0.0f) { z_high = z; sdf_high = smid; }
            z = zpred(sdf_low, sdf_high, z_low, z_high);
        }

        if (net_surface) dists = z;     // secant_mask == sampler_mask & net_surface
        // s_pts == cam + dists*dir in both branches (points[ind] = cam + pts_int[ind]*dir)
        px = fmaf(dists, dx, cx);
        py = fmaf(dists, dy, cy);
        pz = fmaf(dists, dz, cz);
        acc_out = dists;
        net_out = net_surface ? 1.0f : 0.0f;
    }

    // ---- outputs: [pts (M*3)] [net_obj (M)] [acc_s (M)] ----
    float* pout = out + 3ull * (unsigned)i;
    pout[0] = px;
    pout[1] = py;
    pout[2] = pz;
    out[3ull * (unsigned)M + (unsigned)i] = net_out;
    out[4ull * (unsigned)M + (unsigned)i] = acc_out;
}

extern "C" void kernel_launch(void* const* d_in, const int* in_sizes, int n_in,
                              void* d_out, int out_size, void* d_ws, size_t ws_size,
                              hipStream_t stream) {
    const float* cam    = (const float*)d_in[0];   // [1,3]
    const float* dirs   = (const float*)d_in[1];   // [1,N,3]
    // d_in[2] = object_mask : unused by the reference computation
    const float* radius = (const float*)d_in[3];   // [1]

    const int M = in_sizes[1] / 3;
    float* out = (float*)d_out;

    dim3 block(256);
    dim3 grid((M + 255) / 256);
    hipLaunchKernelGGL(raytrace_kernel, grid, block, 0, stream,
                       cam, dirs, radius, out, M);
}